// AttentionalGNN_18983755448761
// MI455X (gfx1250) — compile-verified
//
#include <hip/hip_runtime.h>

// ---------------------------------------------------------------------------
// AttentionalGNN for MI455X (gfx1250), fp32 WMMA (V_WMMA_F32_16X16X4_F32).
// Dead-code-eliminated graph: 3x GAT (sequential on desc3d_db) + final cross
// AttentionPropagation layer only (earlier self/cross outputs are overwritten).
//
// Main GEMM path: block tile 32M x 256N. B tiles are staged into LDS by the
// Tensor Data Mover (TENSOR_LOAD_TO_LDS, one issue per tile from wave 0,
// double-buffered, TENSORcnt + s_wait_tensorcnt), with TDM hardware padding
// producing a bank-conflict-free 264-float LDS pitch. Fragments read via
// ds_load; WMMA with 4 accumulators per wave. Global accesses use uniform
// base + 32-bit offset (SADDR form).
// ---------------------------------------------------------------------------

typedef __attribute__((ext_vector_type(2))) float v2f;
typedef __attribute__((ext_vector_type(8))) float v8f;
typedef __attribute__((ext_vector_type(4))) unsigned int u32x4;
typedef __attribute__((ext_vector_type(8))) int i32x8;
typedef __attribute__((ext_vector_type(4))) int i32x4;

enum {
  F_ATRANS   = 1,    // A[k][m] instead of A[m][k]
  F_BTRANS   = 2,    // B[n][k] instead of B[k][n]  (direct-load path)
  F_BSPLIT   = 4,    // B rows >= splitK come from B2 (fused concat)
  F_BIAS     = 8,    // += bias[row]
  F_ELU1     = 16,   // elu(v)+1
  F_BNRELU   = 32,   // relu(v * g[row]/sqrt(1+eps) + b[row])
  F_RES      = 64,   // += res[row,col]   (residual)
  F_COLSCALE = 128   // *= cs[col] * outScale
};

__device__ __forceinline__ v8f wmma4(v2f a, v2f b, v8f c) {
  return __builtin_amdgcn_wmma_f32_16x16x4_f32(false, a, false, b,
                                               (short)0, c, false, false);
}

__device__ __forceinline__ unsigned int lds_addr_u32(const void* p) {
  return (unsigned int)(uintptr_t)p;  // low 32 bits of LDS flat addr = LDS offset
}

#define LDS_PITCH 264  // 256 cols + 8 pad floats (TDM pad): disjoint bank rows

// One TDM descriptor issue: 2D tile, 16 rows x 256 cols of f32, row stride
// ldb (floats), global -> LDS at lds_off, LDS pitch 264 floats via padding.
// D# packing per CDNA5 ISA.
__device__ __forceinline__ void tdm_load_tile(const float* g, unsigned int lds_off,
                                              int ldb) {
  unsigned long long ga = (unsigned long long)(uintptr_t)g;
  u32x4 g0;
  g0[0] = 1u;                                   // count=1, user mode
  g0[1] = lds_off;                              // lds_addr
  g0[2] = (unsigned int)ga;                     // global_addr[31:0]
  g0[3] = (unsigned int)((ga >> 32) & 0x01FFFFFFull) | (2u << 30);  // addr|type=2
  const unsigned int td0 = 1u << 30, td1 = 1u << 30;  // huge dims: no OOB clamp
  i32x8 g1;
  g1[0] = (int)((2u << 16)    // data_size = 4B
              | (1u << 20)    // pad_enable
              | (7u << 22)    // pad_interval: every 256 DWORDs
              | (7u << 25));  // pad_amount: 8 DWORDs  -> pitch 264 floats
  g1[1] = (int)((td0 & 0xFFFFu) << 16);                   // tensor_dim0[15:0]
  g1[2] = (int)((td0 >> 16) | ((td1 & 0xFFFFu) << 16));   // dim0 hi | dim1 lo
  g1[3] = (int)((td1 >> 16) | (256u << 16));              // dim1 hi | tile_dim0
  g1[4] = 16;                                             // tile_dim1=16, tile_dim2=0
  g1[5] = ldb;                                            // dim0_stride[31:0]
  g1[6] = 0;                                              // dim0_stride hi, dim1_stride lo
  g1[7] = 0;
  i32x4 gz4 = {};
  i32x8 gz8 = {};
  __builtin_amdgcn_tensor_load_to_lds(g0, g1, gz4, gz4, gz8, 0);
}

// C[b] = epilogue( A * B[b] ). Wave tile 16(M) x 64(N), block 32M x 256N.
template <int FLAGS>
__global__ __launch_bounds__(256)
void gemm_wmma_f32(const float* __restrict__ A, const float* __restrict__ B,
                   const float* __restrict__ B2, float* __restrict__ C,
                   const float* __restrict__ bias, const float* __restrict__ bng,
                   const float* __restrict__ bnb, const float* __restrict__ res,
                   const float* __restrict__ cs, float outScale,
                   int M, int N, int K, int lda, int ldb, int ldc, int splitK,
                   long long sA, long long sB, long long sC, long long sRes,
                   long long sCS)
{
  const int tid  = threadIdx.x;
  const int lane = tid & 31;
  const int wave = tid >> 5;                     // 8 waves: 2(M) x 4(N)
  const int wm = wave >> 2, wn = wave & 3;
  const int half = lane >> 4;                    // k-pair {0,1} vs {2,3}
  const int t    = lane & 15;
  const int bz = blockIdx.z;
  const int m0 = blockIdx.y * 32 + wm * 16;
  const int n0 = blockIdx.x * 256 + wn * 64;

  const float* Ab = A + (size_t)bz * sA;

  v8f acc0 = {}, acc1 = {}, acc2 = {}, acc3 = {};

  if constexpr (FLAGS & F_BTRANS) {
    // ---- tiny GEMM path (KV): B[n][k], direct loads, no barriers ----------
    if (m0 >= M || n0 >= N) return;              // wave-uniform exit
    const float* Bb = B + (size_t)bz * sB;
    const int am = m0 + t, bn = n0 + t;
    unsigned int aoff = (unsigned int)(am * lda + half * 2);
    unsigned int b0o = (unsigned int)((bn +  0) * ldb + half * 2);
    unsigned int b1o = (unsigned int)((bn + 16) * ldb + half * 2);
    unsigned int b2o = (unsigned int)((bn + 32) * ldb + half * 2);
    unsigned int b3o = (unsigned int)((bn + 48) * ldb + half * 2);
#pragma unroll 4
    for (int k = 0; k < K; k += 4) {
      v2f a, b0, b1, b2, b3;
      a.x = Ab[aoff]; a.y = Ab[aoff + 1];
      b0.x = Bb[b0o]; b0.y = Bb[b0o + 1];
      b1.x = Bb[b1o]; b1.y = Bb[b1o + 1];
      b2.x = Bb[b2o]; b2.y = Bb[b2o + 1];
      b3.x = Bb[b3o]; b3.y = Bb[b3o + 1];
      acc0 = wmma4(a, b0, acc0);
      acc1 = wmma4(a, b1, acc1);
      acc2 = wmma4(a, b2, acc2);
      acc3 = wmma4(a, b3, acc3);
      aoff += 4; b0o += 4; b1o += 4; b2o += 4; b3o += 4;
    }
  } else {
    // ---- main path: TDM double-buffered B staging in LDS ------------------
    // All launches on this path have M % 32 == 0, N % 256 == 0, K % 16 == 0,
    // so every wave participates (barrier-safe).
    __shared__ float bt[2][16 * LDS_PITCH];
    const float* Bb  = B + (size_t)bz * sB;
    const float* B2b = (FLAGS & F_BSPLIT) ? (B2 + (size_t)bz * sB) : nullptr;
    const int nblk = blockIdx.x * 256;

    auto issue_tile = [&](int kt, int bufi) {
      const int kbase = kt * 16;
      const float* src = Bb;
      int krow = kbase;
      if constexpr (FLAGS & F_BSPLIT) {
        if (kbase >= splitK) { src = B2b; krow = kbase - splitK; }
      }
      tdm_load_tile(src + (size_t)krow * ldb + nblk,
                    lds_addr_u32(&bt[bufi][0]), ldb);
    };

    const int am = m0 + t;
    unsigned int aoff, astep;
    if constexpr (FLAGS & F_ATRANS) {
      aoff = (unsigned int)((half * 2) * lda + am);
      astep = (unsigned int)(4 * lda);
    } else {
      aoff = (unsigned int)(am * lda + half * 2);
      astep = 4;
    }

    const int T = K / 16;
    if (wave == 0) issue_tile(0, 0);             // TDM ignores EXEC; 1 wave issues
    for (int kt = 0; kt < T; ++kt) {
      if (wave == 0) {
        if (kt + 1 < T) {
          issue_tile(kt + 1, (kt + 1) & 1);      // prefetch next tile
          __builtin_amdgcn_s_wait_tensorcnt(1);  // tile kt complete (in-order)
        } else {
          __builtin_amdgcn_s_wait_tensorcnt(0);
        }
      }
      __syncthreads();                           // tile kt visible to all waves
      const float* Lt = &bt[kt & 1][wn * 64 + t];
#pragma unroll
      for (int kk = 0; kk < 4; ++kk) {
        v2f a, b0, b1, b2, b3;
        if constexpr (FLAGS & F_ATRANS) { a.x = Ab[aoff]; a.y = Ab[aoff + lda]; }
        else                            { a.x = Ab[aoff]; a.y = Ab[aoff + 1]; }
        const float* Lr = Lt + (kk * 4 + half * 2) * LDS_PITCH;
        b0.x = Lr[0];  b0.y = Lr[LDS_PITCH + 0];
        b1.x = Lr[16]; b1.y = Lr[LDS_PITCH + 16];
        b2.x = Lr[32]; b2.y = Lr[LDS_PITCH + 32];
        b3.x = Lr[48]; b3.y = Lr[LDS_PITCH + 48];
        acc0 = wmma4(a, b0, acc0);
        acc1 = wmma4(a, b1, acc1);
        acc2 = wmma4(a, b2, acc2);
        acc3 = wmma4(a, b3, acc3);
        aoff += astep;
      }
      __syncthreads();                           // reads done before overwrite
    }
  }

  // ------------------------------ epilogue ---------------------------------
  float* Cb = C + (size_t)bz * sC;
  const float* resb = (FLAGS & F_RES) ? (res + (size_t)bz * sRes) : nullptr;
  const float bnrs = 0.99999500003749971f;  // rsqrt(1 + 1e-5)
  v8f accs[4] = {acc0, acc1, acc2, acc3};
#pragma unroll
  for (int f = 0; f < 4; ++f) {
    const int col = n0 + f * 16 + t;
    float scale = outScale;
    if constexpr (FLAGS & F_COLSCALE) scale *= cs[(size_t)bz * sCS + col];
#pragma unroll
    for (int r = 0; r < 8; ++r) {
      const int row = m0 + r + half * 8;   // C layout: vgpr r -> row r / r+8
      const unsigned int off = (unsigned int)(row * ldc + col);
      float v = accs[f][r] * scale;
      if constexpr (FLAGS & F_BIAS) v += bias[row];
      if constexpr (FLAGS & F_BNRELU) {
        v = v * (bng[row] * bnrs) + bnb[row];
        v = v > 0.f ? v : 0.f;
      }
      if constexpr (FLAGS & F_ELU1) v = v > 0.f ? v + 1.f : __expf(v);
      if constexpr (FLAGS & F_RES) v += resb[off];
      Cb[off] = v;
    }
  }
}

// ksum[b,h,d] = sum_m k[b, d*4+h, m]
__global__ void ksum_kernel(const float* __restrict__ k, float* __restrict__ out, int N) {
  int idx = blockIdx.x * blockDim.x + threadIdx.x;
  if (idx >= 512) return;
  int b = idx >> 8, c = idx & 255;
  int d = c >> 2, h = c & 3;
  const float* p = k + ((size_t)b * 256 + c) * N;
  float s = 0.f;
  for (int m = 0; m < N; ++m) s += p[m];
  out[((size_t)b * 4 + h) * 64 + d] = s;
}

// z[b,h,m] = 1 / (sum_d q[b, d*4+h, m] * ksum[b,h,d] + eps)
__global__ void z_kernel(const float* __restrict__ q, const float* __restrict__ ksum,
                         float* __restrict__ z, int N) {
  int idx = blockIdx.x * blockDim.x + threadIdx.x;
  if (idx >= 8 * N) return;
  int m = idx % N;
  int h = (idx / N) & 3;
  int b = idx / (4 * N);
  const float* qs = q + ((size_t)b * 256 + h) * N + m;
  const float* ks = ksum + ((size_t)b * 4 + h) * 64;
  float s = 0.f;
  for (int d = 0; d < 64; ++d) s += qs[(size_t)d * 4 * N] * ks[d];
  z[idx] = 1.f / (s + 1e-6f);
}

// e[b,n] = sum_o WH[b,o,n] * a[o]
__global__ void gat_e_kernel(const float* __restrict__ WH, const float* __restrict__ a,
                             float* __restrict__ e, int N) {
  int idx = blockIdx.x * blockDim.x + threadIdx.x;
  if (idx >= 2 * N) return;
  int b = idx / N, n = idx % N;
  const float* p = WH + (size_t)b * 256 * N + n;
  float s = 0.f;
  for (int o = 0; o < 256; ++o) s += p[(size_t)o * N] * a[o];
  e[idx] = s;
}

// out[b,o,i] = elu( softmax_j( leaky(e1[i]+e2[16i+j]) ) . WH2[b,o,16i:16i+16] )
__global__ void gat_agg_kernel(const float* __restrict__ WH2, const float* __restrict__ e1,
                               const float* __restrict__ e2, float* __restrict__ out) {
  int idx = blockIdx.x * blockDim.x + threadIdx.x;
  if (idx >= 2 * 256 * 1024) return;
  int i = idx & 1023;
  int o = (idx >> 10) & 255;
  int b = idx >> 18;
  float ei = e1[b * 1024 + i];
  const float* e2p = e2 + b * 16384 + i * 16;
  float w[16];
  float mx = -1e30f;
#pragma unroll
  for (int j = 0; j < 16; ++j) {
    float t = ei + e2p[j];
    t = t > 0.f ? t : 0.2f * t;  // LeakyReLU(0.2)
    w[j] = t;
    mx = fmaxf(mx, t);
  }
  float s = 0.f;
#pragma unroll
  for (int j = 0; j < 16; ++j) { w[j] = __expf(w[j] - mx); s += w[j]; }
  const float inv = 1.f / s;
  const float* p = WH2 + ((size_t)b * 256 + o) * 16384 + i * 16;
  float acc = 0.f;
#pragma unroll
  for (int j = 0; j < 16; ++j) acc += w[j] * p[j];
  acc *= inv;
  out[((size_t)b * 256 + o) * 1024 + i] = acc > 0.f ? acc : __expf(acc) - 1.f;
}

// ---------------------------------------------------------------------------

extern "C" void kernel_launch(void* const* d_in, const int* in_sizes, int n_in,
                              void* d_out, int out_size, void* d_ws, size_t ws_size,
                              hipStream_t stream) {
  constexpr int D = 256, H = 4, NQ = 4096, N3 = 1024, N2 = 16384, B = 2;

  const float* d2q     = (const float*)d_in[0];   // [2,256,4096]
  const float* d3in    = (const float*)d_in[1];   // [2,256,1024]
  const float* d2db    = (const float*)d_in[2];   // [2,256,16384]
  const float* proj_w  = (const float*)d_in[3];   // [6,3,256,256]
  const float* proj_b  = (const float*)d_in[4];   // [6,3,256]
  const float* merge_w = (const float*)d_in[5];   // [6,256,256]
  const float* merge_b = (const float*)d_in[6];   // [6,256]
  const float* mlp1_w  = (const float*)d_in[7];   // [6,512,512]
  const float* mlp1_b  = (const float*)d_in[8];   // [6,512]
  const float* bn_g    = (const float*)d_in[9];   // [6,512]
  const float* bn_b    = (const float*)d_in[10];  // [6,512]
  const float* mlp2_w  = (const float*)d_in[11];  // [6,256,512]
  const float* mlp2_b  = (const float*)d_in[12];  // [6,256]
  const float* gat_W   = (const float*)d_in[13];  // [3,256,256]
  const float* gat_a   = (const float*)d_in[14];  // [3,512]

  float* out0 = (float*)d_out;                    // desc0 [2,256,4096]
  float* out1 = out0 + (size_t)B * D * NQ;        // desc1 [2,256,1024]

  // ---- workspace layout (floats); AP region overlays GAT region -----------
  float* ws  = (float*)d_ws;
  float* d3a = ws;                       // 2*256*1024
  float* d3b = d3a + (size_t)B * D * N3;
  float* big = d3b + (size_t)B * D * N3;
  // GAT phase
  float* WH2 = big;                      // 2*256*16384
  float* WH3 = WH2 + (size_t)B * D * N2; // 2*256*1024
  float* e1  = WH3 + (size_t)B * D * N3; // 2*1024
  float* e2  = e1 + B * N3;              // 2*16384
  // AP phase (reuses `big`)
  float* qb   = big;
  float* kb   = qb  + (size_t)B * D * NQ;
  float* vb   = kb  + (size_t)B * D * NQ;
  float* msgb = vb  + (size_t)B * D * NQ;
  float* mb   = msgb + (size_t)B * D * NQ;
  float* hb   = mb  + (size_t)B * D * NQ;     // 2*512*4096
  float* ksum = hb  + (size_t)B * 2 * D * NQ; // 2*4*64
  float* KV   = ksum + 512;                   // 2*4*64*64
  float* zb   = KV + 2 * 4 * 64 * 64;         // 2*4*4096

  auto gemm = [&](int flags, const float* A, const float* Bm, const float* B2,
                  float* C, const float* bias, const float* bg, const float* bb,
                  const float* res, const float* cs, float oscale,
                  int M, int N, int K, int lda, int ldb, int ldc, int splitK,
                  long long sA, long long sB, long long sC, long long sRes,
                  long long sCS, int nb) {
    dim3 grid((N + 255) / 256, (M + 31) / 32, nb), blk(256);
#define GEMM_CASE(FL)                                                          \
    case (FL):                                                                 \
      gemm_wmma_f32<(FL)><<<grid, blk, 0, stream>>>(                           \
          A, Bm, B2, C, bias, bg, bb, res, cs, oscale, M, N, K, lda, ldb, ldc, \
          splitK, sA, sB, sC, sRes, sCS);                                      \
      break;
    switch (flags) {
      GEMM_CASE(F_ATRANS)
      GEMM_CASE(F_BIAS | F_ELU1)
      GEMM_CASE(F_BIAS)
      GEMM_CASE(F_BTRANS)
      GEMM_CASE(F_COLSCALE)
      GEMM_CASE(F_BIAS | F_BNRELU | F_BSPLIT)
      GEMM_CASE(F_BIAS | F_RES)
      default: break;
    }
#undef GEMM_CASE
  };

  // ======================= 3x GAT layers ===================================
  const float* cur3 = d3in;
  float* nxt3[3] = {d3a, d3b, d3a};
  for (int gi = 0; gi < 3; ++gi) {
    const float* W = gat_W + (size_t)gi * D * D;
    const float* a = gat_a + (size_t)gi * 2 * D;
    // WH2 = W^T * desc2d_db  [256 x 16384], per batch
    gemm(F_ATRANS, W, d2db, nullptr, WH2, nullptr, nullptr, nullptr, nullptr,
         nullptr, 1.f, D, N2, D, D, N2, N2, 0, 0, (long long)D * N2,
         (long long)D * N2, 0, 0, B);
    // WH3 = W^T * desc3d   [256 x 1024]
    gemm(F_ATRANS, W, cur3, nullptr, WH3, nullptr, nullptr, nullptr, nullptr,
         nullptr, 1.f, D, N3, D, D, N3, N3, 0, 0, (long long)D * N3,
         (long long)D * N3, 0, 0, B);
    gat_e_kernel<<<(2 * N3 + 255) / 256, 256, 0, stream>>>(WH3, a, e1, N3);
    gat_e_kernel<<<(2 * N2 + 255) / 256, 256, 0, stream>>>(WH2, a + D, e2, N2);
    gat_agg_kernel<<<(2 * D * N3) / 256, 256, 0, stream>>>(WH2, e1, e2, nxt3[gi]);
    cur3 = nxt3[gi];
  }
  const float* d3f = d3a;  // final desc3d_db (g2 wrote d3a)

  // =============== final cross AttentionPropagation (ai = 5) ===============
  const int ai = 5;
  const float* Pq = proj_w + (size_t)(ai * 3 + 0) * D * D;
  const float* Pk = proj_w + (size_t)(ai * 3 + 1) * D * D;
  const float* Pv = proj_w + (size_t)(ai * 3 + 2) * D * D;
  const float* bq = proj_b + (size_t)(ai * 3 + 0) * D;
  const float* bk = proj_b + (size_t)(ai * 3 + 1) * D;
  const float* bv = proj_b + (size_t)(ai * 3 + 2) * D;
  const float* Wm = merge_w + (size_t)ai * D * D;
  const float* bm = merge_b + (size_t)ai * D;
  const float* W1 = mlp1_w + (size_t)ai * 2 * D * 2 * D;
  const float* b1 = mlp1_b + (size_t)ai * 2 * D;
  const float* g1 = bn_g + (size_t)ai * 2 * D;
  const float* s1 = bn_b + (size_t)ai * 2 * D;
  const float* W2 = mlp2_w + (size_t)ai * D * 2 * D;
  const float* b2 = mlp2_b + (size_t)ai * D;

  auto attn_prop = [&](const float* x, int Nx, const float* src, int Ns, float* out) {
    const long long sx = (long long)D * Nx, ss = (long long)D * Ns;
    // q = elu(Pq x + bq)+1 ;  k = elu(Pk s + bk)+1 ;  v = Pv s + bv
    gemm(F_BIAS | F_ELU1, Pq, x, nullptr, qb, bq, nullptr, nullptr, nullptr,
         nullptr, 1.f, D, Nx, D, D, Nx, Nx, 0, 0, sx, sx, 0, 0, B);
    gemm(F_BIAS | F_ELU1, Pk, src, nullptr, kb, bk, nullptr, nullptr, nullptr,
         nullptr, 1.f, D, Ns, D, D, Ns, Ns, 0, 0, ss, ss, 0, 0, B);
    gemm(F_BIAS, Pv, src, nullptr, vb, bv, nullptr, nullptr, nullptr,
         nullptr, 1.f, D, Ns, D, D, Ns, Ns, 0, 0, ss, ss, 0, 0, B);
    ksum_kernel<<<2, 256, 0, stream>>>(kb, ksum, Ns);
    // per-head KV = (V/Ns) K^T   [64 x 64], contraction length Ns
    for (int h = 0; h < H; ++h) {
      gemm(F_BTRANS, vb + (size_t)h * Ns, kb + (size_t)h * Ns, nullptr,
           KV + (size_t)h * 4096, nullptr, nullptr, nullptr, nullptr, nullptr,
           1.f / (float)Ns, 64, 64, Ns, 4 * Ns, 4 * Ns, 64, 0, ss, ss,
           4 * 4096LL, 0, 0, B);
    }
    z_kernel<<<(8 * Nx + 255) / 256, 256, 0, stream>>>(qb, ksum, zb, Nx);
    // per-head msg = Ns * diag-col(Z) * (KV . Qh)   [64 x Nx]
    for (int h = 0; h < H; ++h) {
      gemm(F_COLSCALE, KV + (size_t)h * 4096, qb + (size_t)h * Nx, nullptr,
           msgb + (size_t)h * Nx, nullptr, nullptr, nullptr, nullptr,
           zb + (size_t)h * Nx, (float)Ns, 64, Nx, 64, 64, 4 * Nx, 4 * Nx, 0,
           4 * 4096LL, sx, sx, 0, (long long)4 * Nx, B);
    }
    // merge
    gemm(F_BIAS, Wm, msgb, nullptr, mb, bm, nullptr, nullptr, nullptr,
         nullptr, 1.f, D, Nx, D, D, Nx, Nx, 0, 0, sx, sx, 0, 0, B);
    // mlp1 on concat([x, msg_merged]) with fused BN+ReLU
    gemm(F_BIAS | F_BNRELU | F_BSPLIT, W1, x, mb, hb, b1, g1, s1, nullptr,
         nullptr, 1.f, 2 * D, Nx, 2 * D, 2 * D, Nx, Nx, D, 0, sx, 2 * sx, 0, 0, B);
    // mlp2 + residual (x) -> out
    gemm(F_BIAS | F_RES, W2, hb, nullptr, out, b2, nullptr, nullptr, x,
         nullptr, 1.f, D, Nx, 2 * D, 2 * D, Nx, Nx, 0, 0, 2 * sx, sx, sx, 0, B);
  };

  // cross layer: desc0 = q + AP(q, d3f) ; desc1 = d3f + AP(d3f, q)
  attn_prop(d2q, NQ, d3f, N3, out0);
  attn_prop(d3f, N3, d2q, NQ, out1);

  (void)in_sizes; (void)n_in; (void)out_size; (void)ws_size;
}